// MaxPool1D_63969242906680
// MI455X (gfx1250) — compile-verified
//
#include <hip/hip_runtime.h>
#include <hip/hip_bf16.h>

typedef __attribute__((ext_vector_type(2))) float v2f;
typedef __attribute__((ext_vector_type(8))) float v8f;

#define NPTS 16384
#define NC   64
#define TOPK 11              // k+1 (includes self, dropped at merge)
#define WPB  8               // waves per block (wave32)

__global__ __launch_bounds__(256) void knn_dgcnn_kernel(
    const float* __restrict__ src,   const float* __restrict__ tgt,
    const float* __restrict__ src_c, const float* __restrict__ tgt_c,
    float* __restrict__ out_all)
{
    const int wave = threadIdx.x >> 5;
    const int lane = threadIdx.x & 31;
    const int n    = lane & 15;          // query column / db row within tile
    const bool hi  = lane >= 16;

    const int prob = blockIdx.y;
    const float* __restrict__ feats  = prob ? tgt   : src;
    const float* __restrict__ coords = prob ? tgt_c : src_c;
    float* __restrict__ out = out_all + (size_t)prob * NPTS * (2 * NC);

    const int qbase = (blockIdx.x * WPB + wave) * 16;

    // ---- B matrix (fixed per wave): col n = (-2qx, -2qy, -2qz, 1) ----
    // 4x16 f32 B layout: lanes 0-15 hold K=0 (v0) / K=1 (v1); lanes 16-31 K=2 / K=3.
    v2f B;
    {
        const int qi = qbase + n;
        const float qx = coords[3 * qi + 0];
        const float qy = coords[3 * qi + 1];
        const float qz = coords[3 * qi + 2];
        B.x = hi ? (-2.0f * qz) : (-2.0f * qx);
        B.y = hi ? 1.0f         : (-2.0f * qy);
    }

    // sorted ascending top-11 (score, db index) per lane
    float lv[TOPK];
    int   li[TOPK];
#pragma unroll
    for (int t = 0; t < TOPK; ++t) { lv[t] = 3.0e38f; li[t] = 0; }

    const int roff_base = hi ? 8 : 0;

    // ---- sweep all db tiles: one K=4 WMMA yields 16x16 selection scores ----
    // score(m, n) = |c_m|^2 - 2 c_m . q_n  (ranking-equivalent to squared distance)
    for (int dbase = 0; dbase < NPTS; dbase += 16) {
        const int m = dbase + n;
        const float cx = coords[3 * m + 0];
        const float cy = coords[3 * m + 1];
        const float cz = coords[3 * m + 2];
        const float sq = fmaf(cz, cz, fmaf(cy, cy, cx * cx));
        // 16x4 f32 A layout: lanes 0-15 row m hold K=0 (v0) / K=1 (v1); lanes 16-31 K=2 / K=3.
        v2f A;
        A.x = hi ? cz : cx;
        A.y = hi ? sq : cy;

        v8f c0 = {};
        v8f d = __builtin_amdgcn_wmma_f32_16x16x4_f32(
            /*neg_a=*/false, A, /*neg_b=*/false, B,
            /*c_mod=*/(short)0, c0, /*reuse_a=*/false, /*reuse_b=*/false);

        // cheap tile-reject: min-tree (7 ops) + one guard instead of 8 guards
        const float m01 = fminf(d[0], d[1]);
        const float m23 = fminf(d[2], d[3]);
        const float m45 = fminf(d[4], d[5]);
        const float m67 = fminf(d[6], d[7]);
        const float tmin = fminf(fminf(m01, m23), fminf(m45, m67));

        if (tmin < lv[TOPK - 1]) {               // rare once lists are warm
            const int jb = dbase + roff_base;
#pragma unroll
            for (int r = 0; r < 8; ++r) {
                const float dv = d[r];
                if (dv < lv[TOPK - 1]) {
                    lv[TOPK - 1] = dv;
                    li[TOPK - 1] = jb + r;
#pragma unroll
                    for (int t = TOPK - 1; t >= 1; --t) {
                        const bool  sw = lv[t] < lv[t - 1];
                        const float va = lv[t - 1], vb = lv[t];
                        const int   ia = li[t - 1], ib = li[t];
                        lv[t - 1] = sw ? vb : va;  lv[t] = sw ? va : vb;
                        li[t - 1] = sw ? ib : ia;  li[t] = sw ? ia : ib;
                    }
                }
            }
        }
    }

    // ---- merge the two half-lists (rows 0-7 half and rows 8-15 half) per query ----
    __shared__ float s_v[WPB][32][TOPK];
    __shared__ int   s_i[WPB][32][TOPK];
#pragma unroll
    for (int t = 0; t < TOPK; ++t) { s_v[wave][lane][t] = lv[t]; s_i[wave][lane][t] = li[t]; }
    __syncthreads();

    // lanes n and n+16 redundantly compute the same merge for query qbase+n
    float mval = -3.0e38f;
    {
        int pa = 0, pb = 0;
#pragma unroll
        for (int t = 0; t < TOPK; ++t) {
            const float av = s_v[wave][n][pa];
            const float bv = s_v[wave][n + 16][pb];
            const bool  ta = av <= bv;
            const int   ix = ta ? s_i[wave][n][pa] : s_i[wave][n + 16][pb];
            pa += ta ? 1 : 0;
            pb += ta ? 0 : 1;
            if (t >= 1) {                          // rank t neighbor supplies channel t-1
                const float f = feats[(size_t)ix * NC + (t - 1)];
                mval = fmaxf(mval, f);
            }
        }
    }

    // ---- coalesced output: row i = [feats | mval - feats] ----
#pragma unroll 1
    for (int q = 0; q < 16; ++q) {
        const float mi = __shfl(mval, q, 32);
        const int i = qbase + q;
        const float f0 = feats[(size_t)i * NC + lane];
        const float f1 = feats[(size_t)i * NC + 32 + lane];
        float* orow = out + (size_t)i * (2 * NC);
        orow[lane]      = f0;
        orow[32 + lane] = f1;
        orow[64 + lane] = mi - f0;
        orow[96 + lane] = mi - f1;
    }
}

extern "C" void kernel_launch(void* const* d_in, const int* in_sizes, int n_in,
                              void* d_out, int out_size, void* d_ws, size_t ws_size,
                              hipStream_t stream) {
    const float* src   = (const float*)d_in[0];
    const float* tgt   = (const float*)d_in[1];
    const float* src_c = (const float*)d_in[2];
    const float* tgt_c = (const float*)d_in[3];
    float* out = (float*)d_out;

    dim3 grid(NPTS / 16 / WPB, 2, 1);   // 128 x 2
    dim3 block(256, 1, 1);              // 8 wave32 waves
    knn_dgcnn_kernel<<<grid, block, 0, stream>>>(src, tgt, src_c, tgt_c, out);
}